// AttnDecoderRNN_4552665333913
// MI455X (gfx1250) — compile-verified
//
#include <hip/hip_runtime.h>
#include <hip/hip_bf16.h>
#include <math.h>

typedef float v2f __attribute__((ext_vector_type(2)));
typedef float v8f __attribute__((ext_vector_type(8)));

#define H 1024
#define L 128
#define V 50257

// ws layout (floats)
#define WS_EMB 0
#define WS_APP 1024
#define WS_X   2048
#define WS_GI  3072
#define WS_GH  6144
#define WS_H0  9216
#define WS_H1  10240
#define WS_RED 11264

__device__ __forceinline__ float waveSum(float v) {
#pragma unroll
  for (int o = 16; o > 0; o >>= 1) v += __shfl_down(v, o, 32);
  return v;
}

// ---- Kernel 1: embedding lookup, attn logits, softmax, attn_applied (1 block)
__global__ void k_prologue(const int* __restrict__ ids,
                           const float* __restrict__ hidden,
                           const float* __restrict__ enc,
                           const float* __restrict__ emb,
                           const float* __restrict__ attn_w,
                           const float* __restrict__ attn_b,
                           float* __restrict__ ws_emb,
                           float* __restrict__ ws_app,
                           float* __restrict__ out_attnw) {
  __shared__ float s_cat[2 * H];
  __shared__ float s_logit[L];
  const int tid = threadIdx.x;
  const int lane = tid & 31, warp = tid >> 5;
  const int token = ids[0];
  for (int i = tid; i < H; i += 256) {
    float e = emb[(size_t)token * H + i];
    ws_emb[i] = e;
    s_cat[i] = e;
    s_cat[H + i] = hidden[i];
  }
  __syncthreads();
  // 8 waves x 16 logits, lane-parallel dot over 2048
  for (int j = warp * 16; j < warp * 16 + 16; ++j) {
    const float* wr = attn_w + (size_t)j * (2 * H);
    float acc = 0.f;
    for (int k = lane; k < 2 * H; k += 32) acc += wr[k] * s_cat[k];
    acc = waveSum(acc);
    if (lane == 0) s_logit[j] = acc + attn_b[j];
  }
  __syncthreads();
  if (tid == 0) {  // softmax over 128, trivially serial
    float m = s_logit[0];
    for (int j = 1; j < L; ++j) m = fmaxf(m, s_logit[j]);
    float s = 0.f;
    for (int j = 0; j < L; ++j) { float e = expf(s_logit[j] - m); s_logit[j] = e; s += e; }
    float inv = 1.0f / s;
    for (int j = 0; j < L; ++j) s_logit[j] *= inv;
  }
  __syncthreads();
  if (tid < L) out_attnw[tid] = s_logit[tid];
  for (int i = tid; i < H; i += 256) {  // attn_applied = w @ enc
    float acc = 0.f;
    for (int l = 0; l < L; ++l) acc += s_logit[l] * enc[(size_t)l * H + i];
    ws_app[i] = acc;
  }
}

// ---- Kernel 2: x = relu(comb_w @ [emb, applied] + b). One wave per row.
__global__ void k_comb(const float* __restrict__ comb_w,
                       const float* __restrict__ comb_b,
                       const float* __restrict__ ws_emb,
                       const float* __restrict__ ws_app,
                       float* __restrict__ ws_x) {
  const int lane = threadIdx.x & 31;
  const int row = (blockIdx.x * blockDim.x + threadIdx.x) >> 5;  // 0..1023
  const float4* w4 = (const float4*)(comb_w + (size_t)row * (2 * H));
  const float4* e4 = (const float4*)ws_emb;
  const float4* a4 = (const float4*)ws_app;
  float acc = 0.f;
#pragma unroll
  for (int j = 0; j < 8; ++j) {
    int it = lane + 32 * j;
    float4 w = w4[it], x = e4[it];
    acc += w.x * x.x + w.y * x.y + w.z * x.z + w.w * x.w;
  }
#pragma unroll
  for (int j = 8; j < 16; ++j) {
    int it = lane + 32 * j;
    float4 w = w4[it], x = a4[it - 256];
    acc += w.x * x.x + w.y * x.y + w.z * x.z + w.w * x.w;
  }
  acc = waveSum(acc);
  if (lane == 0) ws_x[row] = fmaxf(acc + comb_b[row], 0.f);
}

// ---- Kernel 3: GRU gate GEMVs: gi = W_ih@x+b_ih, gh = W_hh@h+b_hh (6144 waves)
__global__ void k_gru_gemv(const float* __restrict__ w_ih,
                           const float* __restrict__ w_hh,
                           const float* __restrict__ b_ih,
                           const float* __restrict__ b_hh,
                           const float* __restrict__ x,
                           const float* __restrict__ h,
                           float* __restrict__ gi, float* __restrict__ gh) {
  const int lane = threadIdx.x & 31;
  const int w = (blockIdx.x * blockDim.x + threadIdx.x) >> 5;  // 0..6143
  const float* mat; const float* vec; const float* bias; float* out; int row;
  if (w < 3 * H) { mat = w_ih; vec = x; bias = b_ih; out = gi; row = w; }
  else           { mat = w_hh; vec = h; bias = b_hh; out = gh; row = w - 3 * H; }
  const float4* m4 = (const float4*)(mat + (size_t)row * H);
  const float4* v4 = (const float4*)vec;
  float acc = 0.f;
#pragma unroll
  for (int j = 0; j < 8; ++j) {
    int it = lane + 32 * j;
    float4 a = m4[it], b = v4[it];
    acc += a.x * b.x + a.y * b.y + a.z * b.z + a.w * b.w;
  }
  acc = waveSum(acc);
  if (lane == 0) out[row] = acc + bias[row];
}

// ---- Kernel 4: GRU elementwise gate math
__global__ void k_gru_gate(const float* __restrict__ gi, const float* __restrict__ gh,
                           const float* __restrict__ hprev,
                           float* __restrict__ hnew_ws, float* __restrict__ hnew_out) {
  int j = blockIdx.x * blockDim.x + threadIdx.x;
  if (j >= H) return;
  float r = 1.f / (1.f + expf(-(gi[j] + gh[j])));
  float z = 1.f / (1.f + expf(-(gi[j + H] + gh[j + H])));
  float n = tanhf(gi[j + 2 * H] + r * gh[j + 2 * H]);
  float hv = (1.f - z) * n + z * hprev[j];
  hnew_ws[j] = hv;
  hnew_out[j] = hv;
}

// ---- Kernel 5: logits = out_w @ h1 + out_b via V_WMMA_F32_16X16X4_F32.
// Each wave owns 16 vocab rows; A = 16x4 out_w tile, B = h1 chunk broadcast
// into all 16 columns, so D[m][*] all hold logit m. 256 accumulating WMMAs.
__global__ void k_logits_wmma(const float* __restrict__ out_w,
                              const float* __restrict__ out_b,
                              const float* __restrict__ h1,
                              float* __restrict__ logits) {
  const int lane = threadIdx.x & 31;
  const int tile = (blockIdx.x * blockDim.x + threadIdx.x) >> 5;
  const int ntiles = (V + 15) / 16;
  if (tile >= ntiles) return;  // wave-uniform; EXEC stays all-1 for WMMA
  const int rowBase = tile * 16;
  int row = rowBase + (lane & 15);
  if (row >= V) row = V - 1;          // clamp (stores are guarded)
  const int koff = (lane >> 4) << 1;  // lanes 0-15 -> k{0,1}; 16-31 -> k{2,3}
  const float* wp = out_w + (size_t)row * H + koff;
  const float* hp = h1 + koff;
  v8f c = {};
  for (int k0 = 0; k0 < H; k0 += 4) {
    v2f a = *(const v2f*)(wp + k0);
    v2f b = *(const v2f*)(hp + k0);
    c = __builtin_amdgcn_wmma_f32_16x16x4_f32(false, a, false, b, (short)0, c,
                                              false, false);
  }
  // D col 0: lanes 0..15 hold M=0..7 in c[0..7]; lanes 16..31 hold M=8..15
  if ((lane & 15) == 0) {
    int base = rowBase + ((lane >> 4) << 3);
#pragma unroll
    for (int m = 0; m < 8; ++m) {
      int r = base + m;
      if (r < V) logits[r] = c[m] + out_b[r];
    }
  }
}

// ---- Kernel 6: max + log-sum-exp over logits (single block, L2 resident)
__global__ void k_reduce(const float* __restrict__ logits, float* __restrict__ red) {
  __shared__ float sm[1024];
  int tid = threadIdx.x;
  float m = -INFINITY;
  for (int i = tid; i < V; i += 1024) m = fmaxf(m, logits[i]);
  sm[tid] = m;
  __syncthreads();
  for (int s = 512; s > 0; s >>= 1) {
    if (tid < s) sm[tid] = fmaxf(sm[tid], sm[tid + s]);
    __syncthreads();
  }
  float gmax = sm[0];
  __syncthreads();
  float s = 0.f;
  for (int i = tid; i < V; i += 1024) s += expf(logits[i] - gmax);
  sm[tid] = s;
  __syncthreads();
  for (int st = 512; st > 0; st >>= 1) {
    if (tid < st) sm[tid] += sm[tid + st];
    __syncthreads();
  }
  if (tid == 0) { red[0] = gmax; red[1] = logf(sm[0]); }
}

// ---- Kernel 7: log_softmax rewrite in place
__global__ void k_lsm(float* __restrict__ logits, const float* __restrict__ red) {
  int i = blockIdx.x * blockDim.x + threadIdx.x;
  if (i < V) logits[i] = logits[i] - red[0] - red[1];
}

extern "C" void kernel_launch(void* const* d_in, const int* in_sizes, int n_in,
                              void* d_out, int out_size, void* d_ws, size_t ws_size,
                              hipStream_t stream) {
  const int*   ids    = (const int*)d_in[0];
  const float* hidden = (const float*)d_in[1];   // (2,1,1024)
  const float* enc    = (const float*)d_in[2];   // (128,1024)
  const float* emb    = (const float*)d_in[3];   // (V,1024)
  const float* attn_w = (const float*)d_in[4];
  const float* attn_b = (const float*)d_in[5];
  const float* comb_w = (const float*)d_in[6];
  const float* comb_b = (const float*)d_in[7];
  const float* w_ih0  = (const float*)d_in[8];
  const float* w_hh0  = (const float*)d_in[9];
  const float* b_ih0  = (const float*)d_in[10];
  const float* b_hh0  = (const float*)d_in[11];
  const float* w_ih1  = (const float*)d_in[12];
  const float* w_hh1  = (const float*)d_in[13];
  const float* b_ih1  = (const float*)d_in[14];
  const float* b_hh1  = (const float*)d_in[15];
  const float* out_w  = (const float*)d_in[16];
  const float* out_b  = (const float*)d_in[17];

  float* out = (float*)d_out;
  float* ws  = (float*)d_ws;

  float* logits    = out;              // [0, V)
  float* out_h0    = out + V;          // new_hidden layer 0
  float* out_h1    = out + V + H;      // new_hidden layer 1
  float* out_attnw = out + V + 2 * H;  // attn_weights (128)

  k_prologue<<<1, 256, 0, stream>>>(ids, hidden, enc, emb, attn_w, attn_b,
                                    ws + WS_EMB, ws + WS_APP, out_attnw);
  k_comb<<<128, 256, 0, stream>>>(comb_w, comb_b, ws + WS_EMB, ws + WS_APP,
                                  ws + WS_X);
  k_gru_gemv<<<768, 256, 0, stream>>>(w_ih0, w_hh0, b_ih0, b_hh0, ws + WS_X,
                                      hidden, ws + WS_GI, ws + WS_GH);
  k_gru_gate<<<4, 256, 0, stream>>>(ws + WS_GI, ws + WS_GH, hidden,
                                    ws + WS_H0, out_h0);
  k_gru_gemv<<<768, 256, 0, stream>>>(w_ih1, w_hh1, b_ih1, b_hh1, ws + WS_H0,
                                      hidden + H, ws + WS_GI, ws + WS_GH);
  k_gru_gate<<<4, 256, 0, stream>>>(ws + WS_GI, ws + WS_GH, hidden + H,
                                    ws + WS_H1, out_h1);

  const int ntiles = (V + 15) / 16;          // 3142 wave-tiles
  const int nblk   = (ntiles + 7) / 8;       // 8 waves / 256-thread block
  k_logits_wmma<<<nblk, 256, 0, stream>>>(out_w, out_b, ws + WS_H1, logits);
  k_reduce<<<1, 1024, 0, stream>>>(logits, ws + WS_RED);
  k_lsm<<<(V + 255) / 256, 256, 0, stream>>>(logits, ws + WS_RED);
}